// Conv_1967095021962
// MI455X (gfx1250) — compile-verified
//
#include <hip/hip_runtime.h>
#include <cstdint>

namespace {

constexpr int N   = 2500;
constexpr int E   = 10000;
constexpr int EMB = 32;
constexpr int H   = 4;
constexpr int DH  = 8;
constexpr int FF  = 32;
constexpr float EPS_LN = 1e-5f;

constexpr int EH  = E * H;          // 40000
constexpr int NH  = N * H;          // 10000
constexpr int WPE = H * FF * EMB;   // 4096 floats per edge (big weights)
constexpr int WEE = H * FF * DH;    // 1024 floats per edge (edge weights)

// ---------------- CDNA5 async global->LDS DMA (ASYNCcnt) ----------------
// Streaming (use-once) weight data: non-temporal hint so the 740MB/call stream
// does not churn L2 lines that have reuse (feat table).
__device__ __forceinline__ void async_b128_nt(uint32_t lds_off, const float* gaddr) {
  asm volatile("global_load_async_to_lds_b128 %0, %1, off th:TH_LOAD_NT"
               :: "v"(lds_off), "v"(gaddr) : "memory");
}
// Reused data (feat rows): default temporal policy.
__device__ __forceinline__ void async_b128(uint32_t lds_off, const float* gaddr) {
  asm volatile("global_load_async_to_lds_b128 %0, %1, off"
               :: "v"(lds_off), "v"(gaddr) : "memory");
}
__device__ __forceinline__ void wait_async0() {
  asm volatile("s_wait_asynccnt 0" ::: "memory");
}
__device__ __forceinline__ uint32_t lds_off_of(const void* p) {
  // generic (flat) address of an LDS object: low 32 bits are the LDS offset
  return (uint32_t)(uintptr_t)p;
}
// copy n4 float4's from global to LDS, strided across the 256-thread block
__device__ __forceinline__ void stage_nt(float* lds, const float* g, int n4, int tid) {
  for (int i = tid; i < n4; i += 256)
    async_b128_nt(lds_off_of(lds + 4 * i), g + 4 * (size_t)i);
}
__device__ __forceinline__ void stage(float* lds, const float* g, int n4, int tid) {
  for (int i = tid; i < n4; i += 256)
    async_b128(lds_off_of(lds + 4 * i), g + 4 * (size_t)i);
}

// float atomic max via monotone int mapping (init memory to -inf)
__device__ __forceinline__ void atomic_max_f32(float* addr, float val) {
  if (val >= 0.0f) atomicMax((int*)addr, __float_as_int(val));
  else             atomicMin((unsigned int*)addr, __float_as_uint(val));
}

// One 2-layer per-edge FFN: sHid = relu(W1*fs + W2*fd + b1 + b2); sO = EW^T*sHid + eb
__device__ __forceinline__ void ffn_phase(
    const float* sW1, const float* sW2, const float* sEW,
    const float* sB1, const float* sB2, const float* sEB,
    const float* sFS, const float* sFD,
    float* sHid, float* sO, int tid)
{
  const int c  = tid & 7;    // 8 partial lanes per output
  const int jr = tid >> 3;   // 32 rows per iteration
  #pragma unroll
  for (int it = 0; it < 4; ++it) {
    const int j = it * 32 + jr;                 // hidden index (h*FF+f), 0..127
    const float4 a = *(const float4*)(sW1 + j * EMB + 4 * c);
    const float4 b = *(const float4*)(sW2 + j * EMB + 4 * c);
    float p = a.x * sFS[4*c+0] + a.y * sFS[4*c+1] + a.z * sFS[4*c+2] + a.w * sFS[4*c+3]
            + b.x * sFD[4*c+0] + b.y * sFD[4*c+1] + b.z * sFD[4*c+2] + b.w * sFD[4*c+3];
    p += __shfl_xor(p, 1, 32);
    p += __shfl_xor(p, 2, 32);
    p += __shfl_xor(p, 4, 32);
    if (c == 0) sHid[j] = fmaxf(p + sB1[j] + sB2[j], 0.0f);
  }
  __syncthreads();
  const int o = tid >> 3;            // output index (h*DH+k), 0..31
  const int h = o >> 3, k = o & 7;
  float p = 0.0f;
  #pragma unroll
  for (int q = 0; q < 4; ++q) {
    const int f = 4 * c + q;
    p += sEW[h * FF * DH + f * DH + k] * sHid[h * FF + f];
  }
  p += __shfl_xor(p, 1, 32);
  p += __shfl_xor(p, 2, 32);
  p += __shfl_xor(p, 4, 32);
  if (c == 0) sO[o] = p + sEB[o];
  __syncthreads();
}

} // namespace

// ---------------- Pass A: per-edge FFNs, logits, segment-max, values ----------------
__global__ __launch_bounds__(256) void edge_kernel(
    const float* __restrict__ feat, const float* __restrict__ query,
    const float* __restrict__ skw, const float* __restrict__ dkw,
    const float* __restrict__ skb, const float* __restrict__ dkb,
    const float* __restrict__ ekw, const float* __restrict__ ekb,
    const float* __restrict__ svw, const float* __restrict__ dvw,
    const float* __restrict__ svb, const float* __restrict__ dvb,
    const float* __restrict__ evw, const float* __restrict__ evb,
    const float* __restrict__ kg,  const float* __restrict__ kbeta,
    const float* __restrict__ vg,  const float* __restrict__ vbeta,
    const int* __restrict__ src,   const int* __restrict__ dst,
    float* __restrict__ logits_ws, float* __restrict__ v_ws,
    float* __restrict__ m_ws)
{
  // key-path buffers
  __shared__ alignas(16) float sKW1[WPE];
  __shared__ alignas(16) float sKW2[WPE];
  __shared__ alignas(16) float sKEW[WEE];
  __shared__ alignas(16) float sKB1[H * FF];
  __shared__ alignas(16) float sKB2[H * FF];
  __shared__ alignas(16) float sKEB[H * DH];
  // value-path buffers (double-buffered so all DMA is issued up-front)
  __shared__ alignas(16) float sVW1[WPE];
  __shared__ alignas(16) float sVW2[WPE];
  __shared__ alignas(16) float sVEW[WEE];
  __shared__ alignas(16) float sVB1[H * FF];
  __shared__ alignas(16) float sVB2[H * FF];
  __shared__ alignas(16) float sVEB[H * DH];
  // shared operands / scratch
  __shared__ alignas(16) float sFS[EMB];
  __shared__ alignas(16) float sFD[EMB];
  __shared__ alignas(16) float sQ[H * DH];
  __shared__ float sHid[H * FF];
  __shared__ float sO[H * DH];

  const int e   = blockIdx.x;
  const int tid = threadIdx.x;
  const int si  = src[e], di = dst[e];

  // ---- issue the entire 72KB per-edge working set as async DMA, one wait ----
  stage_nt(sKW1, skw + (size_t)e * WPE, WPE / 4, tid);
  stage_nt(sKW2, dkw + (size_t)e * WPE, WPE / 4, tid);
  stage_nt(sVW1, svw + (size_t)e * WPE, WPE / 4, tid);
  stage_nt(sVW2, dvw + (size_t)e * WPE, WPE / 4, tid);
  stage_nt(sKEW, ekw + (size_t)e * WEE, WEE / 4, tid);
  stage_nt(sVEW, evw + (size_t)e * WEE, WEE / 4, tid);
  stage_nt(sKB1, skb + (size_t)e * (H * FF), H * FF / 4, tid);
  stage_nt(sKB2, dkb + (size_t)e * (H * FF), H * FF / 4, tid);
  stage_nt(sVB1, svb + (size_t)e * (H * FF), H * FF / 4, tid);
  stage_nt(sVB2, dvb + (size_t)e * (H * FF), H * FF / 4, tid);
  stage_nt(sKEB, ekb + (size_t)e * (H * DH), H * DH / 4, tid);
  stage_nt(sVEB, evb + (size_t)e * (H * DH), H * DH / 4, tid);
  stage   (sFS,  feat + (size_t)si * EMB, EMB / 4, tid);
  stage   (sFD,  feat + (size_t)di * EMB, EMB / 4, tid);
  stage_nt(sQ,   query + (size_t)e * (H * DH), H * DH / 4, tid);
  wait_async0();
  __syncthreads();

  // ---- key path ----
  ffn_phase(sKW1, sKW2, sKEW, sKB1, sKB2, sKEB, sFS, sFD, sHid, sO, tid);

  // per-head LN of k, dot with query -> logits; atomic segment-max over dst
  if (tid < H) {
    float x[DH], mean = 0.0f;
    #pragma unroll
    for (int k = 0; k < DH; ++k) { x[k] = sO[tid * DH + k]; mean += x[k]; }
    mean *= (1.0f / DH);
    float var = 0.0f;
    #pragma unroll
    for (int k = 0; k < DH; ++k) { const float d = x[k] - mean; var += d * d; }
    var *= (1.0f / DH);
    const float inv = rsqrtf(var + EPS_LN);
    float lg = 0.0f;
    #pragma unroll
    for (int k = 0; k < DH; ++k) {
      const float kn = (x[k] - mean) * inv * kg[k] + kbeta[k];
      lg += kn * sQ[tid * DH + k];
    }
    logits_ws[e * H + tid] = lg;
    atomic_max_f32(&m_ws[di * H + tid], lg);
  }
  // NOTE: no extra barrier needed: value ffn_phase only writes sHid (not read
  // by the LN above) before its internal __syncthreads, and writes sO after it.

  // ---- value path ----
  ffn_phase(sVW1, sVW2, sVEW, sVB1, sVB2, sVEB, sFS, sFD, sHid, sO, tid);

  if (tid < H) {
    float x[DH], mean = 0.0f;
    #pragma unroll
    for (int k = 0; k < DH; ++k) { x[k] = sO[tid * DH + k]; mean += x[k]; }
    mean *= (1.0f / DH);
    float var = 0.0f;
    #pragma unroll
    for (int k = 0; k < DH; ++k) { const float d = x[k] - mean; var += d * d; }
    var *= (1.0f / DH);
    const float inv = rsqrtf(var + EPS_LN);
    #pragma unroll
    for (int k = 0; k < DH; ++k)
      v_ws[(size_t)e * (H * DH) + tid * DH + k] =
          (x[k] - mean) * inv * vg[k] + vbeta[k];
  }
}

// ---------------- Pass 0: init segment-max / sums / aggregate ----------------
__global__ void init_kernel(float* __restrict__ m_ws, float* __restrict__ s_ws,
                            float* __restrict__ agg) {
  const int t = blockIdx.x * blockDim.x + threadIdx.x;
  if (t < NH) { m_ws[t] = __int_as_float(0xff800000); s_ws[t] = 0.0f; }
  if (t < NH * DH) agg[t] = 0.0f;
}

// ---------------- Pass B: ex = exp(l - m[dst]); s[dst] += ex ----------------
__global__ void ex_kernel(const float* __restrict__ logits, const float* __restrict__ m_ws,
                          const int* __restrict__ dst,
                          float* __restrict__ ex_ws, float* __restrict__ s_ws) {
  const int t = blockIdx.x * blockDim.x + threadIdx.x;
  if (t >= EH) return;
  const int e = t >> 2, h = t & 3;
  const int d = dst[e];
  const float ex = __expf(logits[t] - m_ws[d * H + h]);
  ex_ws[t] = ex;
  atomicAdd(&s_ws[d * H + h], ex);
}

// ---------------- Pass C: attn = ex/s[dst]; agg[dst] += v*attn ----------------
__global__ void attn_agg_kernel(const float* __restrict__ ex_ws, const float* __restrict__ s_ws,
                                const float* __restrict__ v_ws, const int* __restrict__ dst,
                                float* __restrict__ attn_out, float* __restrict__ agg) {
  const int t = blockIdx.x * blockDim.x + threadIdx.x;
  if (t >= EH) return;
  const int e = t >> 2, h = t & 3;
  const int d = dst[e];
  const float a = ex_ws[t] / s_ws[d * H + h];
  attn_out[t] = a;
  #pragma unroll
  for (int k = 0; k < DH; ++k)
    atomicAdd(&agg[(d * H + h) * DH + k], v_ws[(size_t)e * (H * DH) + h * DH + k] * a);
}

// ---------------- Pass D: final per-head LN over aggregate ----------------
__global__ void out_ln_kernel(const float* __restrict__ agg,
                              const float* __restrict__ og, const float* __restrict__ ob,
                              float* __restrict__ out) {
  const int t = blockIdx.x * blockDim.x + threadIdx.x;   // t = n*H + h
  if (t >= NH) return;
  float x[DH], mean = 0.0f;
  #pragma unroll
  for (int k = 0; k < DH; ++k) { x[k] = agg[t * DH + k]; mean += x[k]; }
  mean *= (1.0f / DH);
  float var = 0.0f;
  #pragma unroll
  for (int k = 0; k < DH; ++k) { const float d = x[k] - mean; var += d * d; }
  var *= (1.0f / DH);
  const float inv = rsqrtf(var + EPS_LN);
  #pragma unroll
  for (int k = 0; k < DH; ++k)
    out[t * DH + k] = (x[k] - mean) * inv * og[k] + ob[k];
}

extern "C" void kernel_launch(void* const* d_in, const int* in_sizes, int n_in,
                              void* d_out, int out_size, void* d_ws, size_t ws_size,
                              hipStream_t stream) {
  (void)in_sizes; (void)n_in; (void)out_size; (void)ws_size;

  const float* feat  = (const float*)d_in[0];
  const float* query = (const float*)d_in[1];
  const float* skw   = (const float*)d_in[2];
  const float* dkw   = (const float*)d_in[3];
  const float* skb   = (const float*)d_in[4];
  const float* dkb   = (const float*)d_in[5];
  const float* ekw   = (const float*)d_in[6];
  const float* ekb   = (const float*)d_in[7];
  const float* svw   = (const float*)d_in[8];
  const float* dvw   = (const float*)d_in[9];
  const float* svb   = (const float*)d_in[10];
  const float* dvb   = (const float*)d_in[11];
  const float* evw   = (const float*)d_in[12];
  const float* evb   = (const float*)d_in[13];
  const float* kg    = (const float*)d_in[14];
  const float* kbeta = (const float*)d_in[15];
  const float* vg    = (const float*)d_in[16];
  const float* vbeta = (const float*)d_in[17];
  const float* og    = (const float*)d_in[18];
  const float* ob    = (const float*)d_in[19];
  const int*   src   = (const int*)d_in[20];
  const int*   dst   = (const int*)d_in[21];
  // d_in[22] is i == 0 (L == 1): all per-layer offsets are zero.

  float* ws     = (float*)d_ws;
  float* logits = ws;                 // EH      = 40000
  float* ex     = ws + 40000;         // EH      = 40000
  float* vws    = ws + 80000;         // E*H*DH  = 320000
  float* m      = ws + 400000;        // NH      = 10000
  float* s      = ws + 410000;        // NH      = 10000
  float* agg    = ws + 420000;        // NH*DH   = 80000  (total 2 MB)

  float* out  = (float*)d_out;        // (N, EMB) = 80000 floats
  float* attn = out + N * EMB;        // (E, H)   = 40000 floats

  init_kernel<<<(NH * DH + 255) / 256, 256, 0, stream>>>(m, s, agg);

  edge_kernel<<<E, 256, 0, stream>>>(
      feat, query, skw, dkw, skb, dkb, ekw, ekb,
      svw, dvw, svb, dvb, evw, evb,
      kg, kbeta, vg, vbeta, src, dst,
      logits, vws, m);

  ex_kernel<<<(EH + 255) / 256, 256, 0, stream>>>(logits, m, dst, ex, s);

  attn_agg_kernel<<<(EH + 255) / 256, 256, 0, stream>>>(ex, s, vws, dst, attn, agg);

  out_ln_kernel<<<(NH + 255) / 256, 256, 0, stream>>>(agg, og, ob, out);
}